// Expm_54872502174211
// MI455X (gfx1250) — compile-verified
//
#include <hip/hip_runtime.h>
#include <math.h>

// ---------------------------------------------------------------------------
// expm of per-voxel symmetric 3x3, layout (B, 9, D*H*W), f32.
// Closed form: Smith trigonometric eigenvalues + Newton divided differences:
//   exp(S) = f0*I + f1*(S - l0 I) + f2*(S^2 - (l0+l1) S + l0 l1 I)
// Input is symmetric by construction -> load only 6 of 9 channels.
// Data path: 1024-voxel tiles staged to LDS via global_load_async_to_lds_b128,
// double buffered; each block owns 3 contiguous tiles (never straddles a batch
// element since 96^3 = 3072*288), so batch index is one div per block.
// ---------------------------------------------------------------------------

typedef float f4  __attribute__((ext_vector_type(4)));
typedef int   v4i __attribute__((ext_vector_type(4)));

namespace {
constexpr int  kNvox     = 96 * 96 * 96;          // 884736 voxels per batch elem
constexpr int  kTile     = 1024;                  // voxels per tile
constexpr int  kThreads  = kTile / 4;             // 256 threads, 4 voxels each
constexpr int  kTilesPB  = 3;                     // contiguous tiles per block
constexpr int  kBlocksPerBatch = kNvox / (kTile * kTilesPB);  // 288
constexpr int  kGrid     = 4 * kBlocksPerBatch;   // 1152 blocks (B = 4)
}

#if defined(__HIP_DEVICE_COMPILE__) && \
    __has_builtin(__builtin_amdgcn_global_load_async_to_lds_b128)
#define EXPM_ASYNC 1
#else
#define EXPM_ASYNC 0
#endif

#if EXPM_ASYNC
typedef __attribute__((address_space(1))) v4i gv4i;  // global b128
typedef __attribute__((address_space(3))) v4i lv4i;  // LDS b128

template <int N>
__device__ __forceinline__ void wait_async() {
#if __has_builtin(__builtin_amdgcn_s_wait_asynccnt)
  __builtin_amdgcn_s_wait_asynccnt(N);
#else
  asm volatile("s_wait_asynccnt %0" ::"i"(N) : "memory");
#endif
  asm volatile("" ::: "memory");  // keep LDS reads below the wait
}
#endif

// e[6] = {e00, e01, e02, e11, e12, e22}
__device__ __forceinline__ void expm3(float a00, float a01, float a02,
                                      float a11, float a12, float a22,
                                      float* __restrict__ e) {
  const float inv3 = 0.33333333333333f;
  float q   = (a00 + a11 + a22) * inv3;
  float b00 = a00 - q, b11 = a11 - q, b22 = a22 - q;
  float off = a01 * a01 + a02 * a02 + a12 * a12;
  float p2  = b00 * b00 + b11 * b11 + b22 * b22 + 2.0f * off;
  float p   = __builtin_sqrtf(p2 * (1.0f / 6.0f));
  float pc  = fmaxf(p, 1e-20f);
  // det(S - qI), row-0 cofactor expansion
  float det = b00 * (b11 * b22 - a12 * a12)
            - a01 * (a01 * b22 - a12 * a02)
            + a02 * (a01 * a12 - b11 * a02);
  float r = 0.5f * det / (pc * pc * pc);
  r = fminf(1.0f, fmaxf(-1.0f, r));
  float phi = acosf(r) * inv3;                 // phi in [0, pi/3]
  float c0  = __cosf(phi);
  float c2  = __cosf(phi + 2.0943951023932f);  // + 2*pi/3
  float l0  = q + 2.0f * p * c0;               // largest
  float l2  = q + 2.0f * p * c2;               // smallest
  float l1  = 3.0f * q - l0 - l2;              // l0 >= l1 >= l2

  float E0 = __expf(l0), E1 = __expf(l1), E2 = __expf(l2);
  float d01 = l0 - l1, d12 = l1 - l2, d02 = l0 - l2;  // all >= 0
  const float tdd = 0.02f;  // cancellation guard for divided differences
  float dd01 = (d01 > tdd) ? (E0 - E1) / d01 : __builtin_sqrtf(E0 * E1);
  float dd12 = (d12 > tdd) ? (E1 - E2) / d12 : __builtin_sqrtf(E1 * E2);
  float f2   = (d02 > tdd) ? (dd01 - dd12) / d02 : 0.5f * E1;
  float f1 = dd01, f0 = E0;

  // S^2 (symmetric, 6 unique entries)
  float s00 = a00 * a00 + a01 * a01 + a02 * a02;
  float s01 = a00 * a01 + a01 * a11 + a02 * a12;
  float s02 = a00 * a02 + a01 * a12 + a02 * a22;
  float s11 = a01 * a01 + a11 * a11 + a12 * a12;
  float s12 = a01 * a02 + a11 * a12 + a12 * a22;
  float s22 = a02 * a02 + a12 * a12 + a22 * a22;

  float kI = f0 - f1 * l0 + f2 * (l0 * l1);  // coefficient on I
  float kS = f1 - f2 * (l0 + l1);            // coefficient on S
  e[0] = kI + kS * a00 + f2 * s00;
  e[1] =      kS * a01 + f2 * s01;
  e[2] =      kS * a02 + f2 * s02;
  e[3] = kI + kS * a11 + f2 * s11;
  e[4] =      kS * a12 + f2 * s12;
  e[5] = kI + kS * a22 + f2 * s22;
}

__device__ __forceinline__ void process_store(f4 A00, f4 A01, f4 A02,
                                              f4 A11, f4 A12, f4 A22,
                                              float* __restrict__ yb) {
  f4 O0, O1, O2, O3, O4, O5;
#pragma unroll
  for (int k = 0; k < 4; ++k) {
    float e[6];
    expm3(A00[k], A01[k], A02[k], A11[k], A12[k], A22[k], e);
    O0[k] = e[0]; O1[k] = e[1]; O2[k] = e[2];
    O3[k] = e[3]; O4[k] = e[4]; O5[k] = e[5];
  }
  // output channels (row-major 3x3): 0:e00 1:e01 2:e02 3:e01 4:e11 5:e12 6:e02 7:e12 8:e22
  __builtin_nontemporal_store(O0, (f4*)(yb + 0L * kNvox));
  __builtin_nontemporal_store(O1, (f4*)(yb + 1L * kNvox));
  __builtin_nontemporal_store(O2, (f4*)(yb + 2L * kNvox));
  __builtin_nontemporal_store(O1, (f4*)(yb + 3L * kNvox));
  __builtin_nontemporal_store(O3, (f4*)(yb + 4L * kNvox));
  __builtin_nontemporal_store(O4, (f4*)(yb + 5L * kNvox));
  __builtin_nontemporal_store(O2, (f4*)(yb + 6L * kNvox));
  __builtin_nontemporal_store(O4, (f4*)(yb + 7L * kNvox));
  __builtin_nontemporal_store(O5, (f4*)(yb + 8L * kNvox));
}

__global__ __launch_bounds__(kThreads) void expm_tile_kernel(
    const float* __restrict__ x, float* __restrict__ y) {
  const int t   = threadIdx.x;
  const int bid = blockIdx.x;
  const int b   = bid / kBlocksPerBatch;                 // batch element
  const int v0b = (bid - b * kBlocksPerBatch) * (kTile * kTilesPB);
  const float* xb = x + (size_t)b * 9 * kNvox + v0b + t * 4;
  float*       yb = y + (size_t)b * 9 * kNvox + v0b + t * 4;

#if EXPM_ASYNC
  __shared__ __align__(16) float lds[2][6][kTile];  // 48 KB
  const int ch[6] = {0, 1, 2, 4, 5, 8};             // unique symmetric channels

  auto issue = [&](int buf, int i) {
#pragma unroll
    for (int s = 0; s < 6; ++s) {
      const float* g = xb + (size_t)ch[s] * kNvox + i * kTile;
      __builtin_amdgcn_global_load_async_to_lds_b128(
          (gv4i*)(void*)g,
          (lv4i*)(void*)&lds[buf][s][t * 4],
          /*offset=*/0, /*cpol=*/0);
    }
  };

  issue(0, 0);
  int cur = 0;
#pragma unroll
  for (int i = 0; i < kTilesPB; ++i) {
    if (i + 1 < kTilesPB) {
      issue(cur ^ 1, i + 1);  // prefetch next tile into the other buffer
      wait_async<6>();        // wait for current tile's 6 loads (in-order)
    } else {
      wait_async<0>();
    }
    f4 A00 = *(const f4*)&lds[cur][0][t * 4];
    f4 A01 = *(const f4*)&lds[cur][1][t * 4];
    f4 A02 = *(const f4*)&lds[cur][2][t * 4];
    f4 A11 = *(const f4*)&lds[cur][3][t * 4];
    f4 A12 = *(const f4*)&lds[cur][4][t * 4];
    f4 A22 = *(const f4*)&lds[cur][5][t * 4];
    process_store(A00, A01, A02, A11, A12, A22, yb + i * kTile);
    cur ^= 1;
  }
#else
#pragma unroll
  for (int i = 0; i < kTilesPB; ++i) {
    const float* g0 = xb + i * kTile;
    f4 A00 = __builtin_nontemporal_load((const f4*)(g0 + 0L * kNvox));
    f4 A01 = __builtin_nontemporal_load((const f4*)(g0 + 1L * kNvox));
    f4 A02 = __builtin_nontemporal_load((const f4*)(g0 + 2L * kNvox));
    f4 A11 = __builtin_nontemporal_load((const f4*)(g0 + 4L * kNvox));
    f4 A12 = __builtin_nontemporal_load((const f4*)(g0 + 5L * kNvox));
    f4 A22 = __builtin_nontemporal_load((const f4*)(g0 + 8L * kNvox));
    process_store(A00, A01, A02, A11, A12, A22, yb + i * kTile);
  }
#endif
}

extern "C" void kernel_launch(void* const* d_in, const int* in_sizes, int n_in,
                              void* d_out, int out_size, void* d_ws,
                              size_t ws_size, hipStream_t stream) {
  (void)in_sizes; (void)n_in; (void)out_size; (void)d_ws; (void)ws_size;
  const float* x = (const float*)d_in[0];
  float* y = (float*)d_out;
  expm_tile_kernel<<<dim3(kGrid), dim3(kThreads), 0, stream>>>(x, y);
}